// CalculateSLayer_32787780338124
// MI455X (gfx1250) — compile-verified
//
#include <hip/hip_runtime.h>
#include <hip/hip_bf16.h>
#include <math.h>

// Problem constants (from reference)
#define NDOC   1024   // N
#define DEP    10
#define H2     60
#define DOUT   70     // DEP + H2
#define NTYPE  50
#define ITILE  4      // i-rows per gather block

#define KPAD   64     // padded K stride for transposed W (60 -> 64)
#define FPAD   80     // padded F extent for transposed W (70 -> 80)

typedef float v2f __attribute__((ext_vector_type(2)));
typedef float v8f __attribute__((ext_vector_type(8)));

// ---------------------------------------------------------------------------
// K1 (prep): E[t,f] = sum_d emb[t,d] * W[60+d, f]   (50x70)
//            WhT[f][k] = W[k][f] zero-padded to [80][64] for branch-free
//            b64 B-operand loads in the WMMA kernel.
// ---------------------------------------------------------------------------
__global__ void prep_kernel(const float* __restrict__ emb,
                            const float* __restrict__ W,
                            float* __restrict__ E,
                            float* __restrict__ WhT) {
    for (int idx = threadIdx.x; idx < NTYPE * DOUT; idx += blockDim.x) {
        int t = idx / DOUT, f = idx % DOUT;
        float a = 0.f;
#pragma unroll
        for (int d = 0; d < DEP; ++d)
            a += emb[t * DEP + d] * W[(H2 + d) * DOUT + f];
        E[idx] = a;
    }
    for (int idx = threadIdx.x; idx < FPAD * KPAD; idx += blockDim.x) {
        int f = idx / KPAD, k = idx % KPAD;
        WhT[idx] = (f < DOUT && k < H2) ? W[k * DOUT + f] : 0.f;
    }
}

// ---------------------------------------------------------------------------
// K2: hW = h[1024,60] @ W[0:60, 0:70] via V_WMMA_F32_16X16X4_F32.
// One wave (32 threads) per 16x16 output tile. grid = (64 i-tiles, 5 f-tiles).
// A 16x4 layout: lane m=tid&15 is row M; half=tid>>4 selects K-pair {0,1}/{2,3}.
// B 4x16 layout: lane n=tid&15 is col N; same half split over K rows.
//   B fetched from zero-padded transposed WhT -> unconditional b64 loads.
// C/D 16x16: VGPR r holds row M = r + 8*half, col N = lane&15.
// ---------------------------------------------------------------------------
__global__ __launch_bounds__(32) void hw_wmma_kernel(const float* __restrict__ h,
                                                     const float* __restrict__ WhT,
                                                     float* __restrict__ hW) {
    const int tid  = threadIdx.x;
    const int m    = tid & 15;
    const int half = tid >> 4;
    const int klo  = half * 2;
    const int i0   = blockIdx.x * 16;
    const int f0   = blockIdx.y * 16;

    const int f = f0 + m;                 // 0..79 (cols >=70 are zero-padded)
    const float* arow = h   + (i0 + m) * H2;
    const float* brow = WhT + f * KPAD;

    v8f c = {};
#pragma unroll
    for (int ks = 0; ks < H2 / 4; ++ks) {
        const int k = ks * 4 + klo;
        v2f a  = *(const v2f*)(arow + k);     // h[i][k], h[i][k+1]
        v2f bb = *(const v2f*)(brow + k);     // W[k][f], W[k+1][f]
        c = __builtin_amdgcn_wmma_f32_16x16x4_f32(
                /*neg_a=*/false, a, /*neg_b=*/false, bb,
                /*c_mod=*/(short)0, c, /*reuse_a=*/false, /*reuse_b=*/false);
    }

    if (f < DOUT) {
#pragma unroll
        for (int r = 0; r < 8; ++r) {
            const int i = i0 + r + 8 * half;
            hW[i * DOUT + f] = c[r];
        }
    }
}

// ---------------------------------------------------------------------------
// K3: zero the s_out half of d_out (accumulated with global atomics in K4).
// ---------------------------------------------------------------------------
__global__ void zero_kernel(float* __restrict__ p, int n) {
    int i = blockIdx.x * blockDim.x + threadIdx.x;
    if (i < n) p[i] = 0.f;
}

// ---------------------------------------------------------------------------
// K4: per-block gather over ITILE i-rows.
//  Phase A: LDS tanh table Tl[il][ty*70+f] = tanh(hW[i,f] + b[f] + E[ty,f])
//  Phase B: type histogram per i (LDS atomics)  -> s_in via weighted sum
//  Phase C: scan all j: 3 j's x 70 f-lanes per step; (matrix,mask) k-pairs
//           loaded as int2 (b64); register-accumulate the ITILE partial,
//           then one global atomicAdd per (j,f). Light prefetch 24 j ahead.
// ---------------------------------------------------------------------------
__global__ __launch_bounds__(256) void s_gather_kernel(
        const float* __restrict__ Eg, const float* __restrict__ hWg,
        const float* __restrict__ bg,
        const int* __restrict__ matrix, const int* __restrict__ mask,
        float* __restrict__ s_in, float* __restrict__ s_out) {
    __shared__ float Tl[ITILE][NTYPE * DOUT];   // 56000 B
    __shared__ float hwb[ITILE][DOUT];          //  1120 B
    __shared__ int   cnt[ITILE][64];            //  1024 B

    const int tid = threadIdx.x;
    const int i0  = blockIdx.x * ITILE;

    const int2* mat2 = (const int2*)matrix;     // [N, N] of (k0,k1)
    const int2* msk2 = (const int2*)mask;

    // hW row + bias into LDS; zero histograms
    for (int idx = tid; idx < ITILE * DOUT; idx += 256) {
        int il = idx / DOUT, f = idx % DOUT;
        hwb[il][f] = hWg[(i0 + il) * DOUT + f] + bg[f];
    }
    if (tid < ITILE * 64) cnt[tid >> 6][tid & 63] = 0;
    __syncthreads();

    // tanh table (recomputed per block: cheaper than 28MB of global traffic)
    for (int idx = tid; idx < ITILE * NTYPE * DOUT; idx += 256) {
        int il = idx / (NTYPE * DOUT);
        int r  = idx % (NTYPE * DOUT);
        int f  = r % DOUT;
        Tl[il][r] = tanhf(hwb[il][f] + Eg[r]);
    }

    // type histogram over (j,k) for each i in the tile
    for (int idx = tid; idx < ITILE * NDOC; idx += 256) {
        int il = idx / NDOC;
        int j  = idx % NDOC;
        long g = (long)(i0 + il) * NDOC + j;
        int2 mm = mat2[g];
        int2 kk = msk2[g];
        if (kk.x) atomicAdd(&cnt[il][mm.x], 1);
        if (kk.y) atomicAdd(&cnt[il][mm.y], 1);
    }
    __syncthreads();

    // s_in[i,f] = sum_ty cnt[i,ty] * T[i,ty,f]
    for (int idx = tid; idx < ITILE * DOUT; idx += 256) {
        int il = idx / DOUT, f = idx % DOUT;
        float acc = 0.f;
#pragma unroll 10
        for (int ty = 0; ty < NTYPE; ++ty)
            acc += (float)cnt[il][ty] * Tl[il][ty * DOUT + f];
        s_in[(i0 + il) * DOUT + f] = acc;
    }

    // s_out: scan j, gather from LDS tanh table, one atomic per (j,f)
    const int JPJ = 3;                       // 3*70 = 210 active lanes
    if (tid < JPJ * DOUT) {
        const int jl = tid / DOUT, f = tid % DOUT;
        for (int jj = 0; jj < NDOC; jj += JPJ) {
            const int j = jj + jl;
            if (j < NDOC) {
                // one lane per j-group pulls lines ~24 j ahead into cache
                if (f == 0 && j + 24 < NDOC) {
                    __builtin_prefetch(&mat2[(long)i0 * NDOC + j + 24], 0, 0);
                    __builtin_prefetch(&msk2[(long)i0 * NDOC + j + 24], 0, 0);
                }
                float acc = 0.f;
#pragma unroll
                for (int il = 0; il < ITILE; ++il) {
                    long g  = (long)(i0 + il) * NDOC + j;
                    int2 mm = mat2[g];
                    int2 kk = msk2[g];
                    if (kk.x) acc += Tl[il][mm.x * DOUT + f];
                    if (kk.y) acc += Tl[il][mm.y * DOUT + f];
                }
                atomicAdd(&s_out[j * DOUT + f], acc);
            }
        }
    }
}

// ---------------------------------------------------------------------------
extern "C" void kernel_launch(void* const* d_in, const int* in_sizes, int n_in,
                              void* d_out, int out_size, void* d_ws, size_t ws_size,
                              hipStream_t stream) {
    const float* h      = (const float*)d_in[0];   // [1024,60]
    const float* emb    = (const float*)d_in[1];   // [50,10]
    const float* W      = (const float*)d_in[2];   // [70,70]
    const float* b      = (const float*)d_in[3];   // [70]
    const int*   matrix = (const int*)d_in[4];     // [1024,1024,2]
    const int*   mask   = (const int*)d_in[5];     // [1024,1024,2]

    float* out   = (float*)d_out;
    float* s_in  = out;                        // [1024,70]
    float* s_out = out + NDOC * DOUT;          // [1024,70]

    float* E   = (float*)d_ws;                 // 3500 floats (slot: 4096)
    float* WhT = E + 4096;                     // 5120 floats (slot: 6144)
    float* hW  = WhT + 6144;                   // 71680 floats

    prep_kernel<<<1, 256, 0, stream>>>(emb, W, E, WhT);
    hw_wmma_kernel<<<dim3(NDOC / 16, (DOUT + 15) / 16), 32, 0, stream>>>(h, WhT, hW);
    zero_kernel<<<(NDOC * DOUT + 255) / 256, 256, 0, stream>>>(s_out, NDOC * DOUT);
    s_gather_kernel<<<NDOC / ITILE, 256, 0, stream>>>(E, hW, b, matrix, mask,
                                                      s_in, s_out);
}